// LIF_80504866996738
// MI455X (gfx1250) — compile-verified
//
#include <hip/hip_runtime.h>
#include <cstdint>
#include <cstddef>

// LIF scan: u_t = decay*u_{t-1} + x_t - o_{t-1}*Vth ; o_t = H(u_t - Vth)
// x: [B=64, N=4096, T=100] f32, T contiguous. 262,144 independent rows.
// Bandwidth-bound: ~210MB total traffic -> ~9us floor at 23.3 TB/s.

#define T_LEN   100
#define CHUNK   20          // timesteps per tile (5 x float4)
#define NCHUNK  5           // 5 * 20 = 100
#define ROWS_PB 256         // rows per block == blockDim.x (8 wave32s)
#define VTH     0.5f

#if __has_builtin(__builtin_amdgcn_tensor_load_to_lds) && __has_builtin(__builtin_amdgcn_s_wait_tensorcnt)
#define USE_TDM 1
#else
#define USE_TDM 0
#endif

typedef __attribute__((ext_vector_type(4))) unsigned int u32x4;
typedef __attribute__((ext_vector_type(8))) int          i32x8;
typedef __attribute__((ext_vector_type(4))) int          i32x4;

#if USE_TDM
typedef __attribute__((address_space(3))) const void lds_cvoid_t;

// Byte offset of a generic (__shared__) pointer within the workgroup LDS allocation.
__device__ __forceinline__ unsigned lif_lds_offset(const void* p) {
  return (unsigned)(size_t)(lds_cvoid_t*)p;
}

// Issue one TDM 2D tile copy: ROWS_PB rows x CHUNK floats, row stride T_LEN floats,
// packed contiguously into LDS at lds_byte (row stride CHUNK*4 = 80B).
// D# bitfields per CDNA5 ISA ch.8 (group0: count/lds_addr/global_addr/type,
// group1: data_size/dims/strides). Groups 2/3 zero => tile_dim2/3 unused (<=2D).
__device__ __forceinline__ void lif_tdm_load_tile(const float* gsrc, unsigned lds_byte) {
  const unsigned long long ga = (unsigned long long)(uintptr_t)gsrc;
  u32x4 g0;
  g0[0] = 1u;                                                  // count=1, user descriptor
  g0[1] = lds_byte;                                            // lds_addr (bytes)
  g0[2] = (unsigned)(ga & 0xFFFFFFFFull);                      // global_addr[31:0]
  g0[3] = ((unsigned)(ga >> 32) & 0x01FFFFFFu) | (2u << 30);   // global_addr[56:32] | type=2
  i32x8 g1;
  g1[0] = (int)(2u << 16);                                     // data_size=2 (4 bytes), wg_mask=0
  g1[1] = (int)((unsigned)CHUNK << 16);                        // tensor_dim0[15:0] @ bits[63:48]
  g1[2] = (int)((unsigned)ROWS_PB << 16);                      // tensor_dim1[15:0] @ bits[95:80]
  g1[3] = (int)((unsigned)CHUNK << 16);                        // tile_dim0 @ bits[127:112]
  g1[4] = (int)ROWS_PB;                                        // tile_dim1 @ [143:128]; tile_dim2=0
  g1[5] = (int)T_LEN;                                          // tensor_dim0_stride[31:0] (elements)
  g1[6] = 0;                                                   // stride0 hi, dim1_stride lo
  g1[7] = 0;
  i32x4 z4 = {0, 0, 0, 0};
#if defined(__clang_major__) && (__clang_major__ >= 23)
  i32x8 z8 = {0, 0, 0, 0, 0, 0, 0, 0};
  __builtin_amdgcn_tensor_load_to_lds(g0, g1, z4, z4, z8, 0);
#else
  __builtin_amdgcn_tensor_load_to_lds(g0, g1, z4, z4, 0);
#endif
}
#endif  // USE_TDM

// 4 serial LIF steps. ov carries o_{t-1}*Vth (pre-scaled spike).
__device__ __forceinline__ void lif_step4(const float4 v, float4& o4,
                                          float& u, float& ov, const float decay) {
  const float* xi = &v.x;
  float* oo = &o4.x;
#pragma unroll
  for (int c = 0; c < 4; ++c) {
    u = __builtin_fmaf(decay, u, xi[c] - ov);  // decay*u + x_t - o_prev*Vth
    const bool s = (u > VTH);                  // H(u - Vth)
    oo[c] = s ? 1.0f : 0.0f;
    ov    = s ? VTH  : 0.0f;
  }
}

__global__ __launch_bounds__(ROWS_PB)
void lif_scan_kernel(const float* __restrict__ x,
                     const float* __restrict__ decay_ptr,
                     float* __restrict__ out,
                     int nrows) {
  const float decay = decay_ptr[0];
  const int tid    = (int)threadIdx.x;
  const int rowblk = (int)blockIdx.x * ROWS_PB;
  const int row    = rowblk + tid;

  float u = 0.0f, ov = 0.0f;
  float* orow = out + (size_t)row * T_LEN;

#if USE_TDM
  __shared__ float smem[2][ROWS_PB * CHUNK];   // 2 x 20KB double buffer
  const float* gtile = x + (size_t)rowblk * T_LEN;
  const unsigned lds0 = lif_lds_offset(&smem[0][0]);
  const unsigned lds1 = lif_lds_offset(&smem[1][0]);
  const bool issuer = (tid < 32);              // wave 0 drives the TDM pipeline

  if (issuer) lif_tdm_load_tile(gtile, lds0);  // prologue: chunk 0 in flight

#pragma unroll
  for (int k = 0; k < NCHUNK; ++k) {
    if (issuer) {
      if (k + 1 < NCHUNK) {
        lif_tdm_load_tile(gtile + (size_t)(k + 1) * CHUNK,
                          ((k + 1) & 1) ? lds1 : lds0);
        __builtin_amdgcn_s_wait_tensorcnt((short)1);   // chunk k landed; k+1 in flight
      } else {
        __builtin_amdgcn_s_wait_tensorcnt((short)0);   // last chunk landed
      }
    }
    __syncthreads();                            // publish LDS tile to all 8 waves
    if (row < nrows) {
      // Row stride = 20 floats (80B): lanes 0..15 of a b128 read cover all 64
      // banks exactly once -> conflict-free ds_load_b128.
      const float4* rp = (const float4*)&smem[k & 1][tid * CHUNK];
      float4* op = (float4*)(orow + k * CHUNK);
#pragma unroll
      for (int j = 0; j < CHUNK / 4; ++j) {
        float4 o4;
        const float4 v = rp[j];
        lif_step4(v, o4, u, ov, decay);
        op[j] = o4;
      }
    }
    __syncthreads();                            // tile consumed; safe to overwrite at k+2
  }
#else
  // Fallback: per-thread row streaming, fully unrolled float4 loads (L2-backed).
  if (row < nrows) {
    const float4* gx = (const float4*)(x + (size_t)row * T_LEN);
    float4* op = (float4*)orow;
#pragma unroll
    for (int j = 0; j < T_LEN / 4; ++j) {
      float4 o4;
      const float4 v = gx[j];
      lif_step4(v, o4, u, ov, decay);
      op[j] = o4;
    }
  }
#endif
}

extern "C" void kernel_launch(void* const* d_in, const int* in_sizes, int n_in,
                              void* d_out, int out_size, void* d_ws, size_t ws_size,
                              hipStream_t stream) {
  (void)n_in; (void)out_size; (void)d_ws; (void)ws_size;
  const float* x     = (const float*)d_in[0];
  const float* decay = (const float*)d_in[1];
  float* out         = (float*)d_out;

  const int total  = in_sizes[0];          // B*N*T = 26,214,400
  const int nrows  = total / T_LEN;        // 262,144
  const int blocks = (nrows + ROWS_PB - 1) / ROWS_PB;  // 1024

  lif_scan_kernel<<<dim3(blocks), dim3(ROWS_PB), 0, stream>>>(x, decay, out, nrows);
}